// MorphModule_36919538876698
// MI455X (gfx1250) — compile-verified
//
#include <hip/hip_runtime.h>

#define NEGV (-10000.0f)

// gfx1250 async global->LDS path (ASYNCcnt). Guarded so the kernel still
// compiles (with a synchronous staging fallback) if the builtin is absent.
#if __has_builtin(__builtin_amdgcn_global_load_async_to_lds_b128)
#define HAVE_ASYNC_LDS 1
#endif

// Exact parameter types per the compiler diagnostic:
//   param0: __attribute__((vector_size(16))) int  addrspace(1)*   (global)
//   param1: expected symmetric LDS pointer        addrspace(3)*   (shared)
typedef int v4i_t __attribute__((vector_size(16)));
typedef __attribute__((address_space(1))) v4i_t* gv4_p;
typedef __attribute__((address_space(3))) v4i_t* lv4_p;

__device__ __forceinline__ void wait_async_zero() {
#if defined(HAVE_ASYNC_LDS)
#if __has_builtin(__builtin_amdgcn_s_wait_asynccnt)
    __builtin_amdgcn_s_wait_asynccnt(0);
#else
    asm volatile("s_wait_asynccnt 0" ::: "memory");
#endif
#endif
}

// Fused double 3x3 grayscale erosion, exact reference semantics (including
// the -10000 re-padding of the intermediate at image borders).
__global__ __launch_bounds__(256) void erode2_fused(
    const float* __restrict__ x,
    const float* __restrict__ s1g,
    const float* __restrict__ s2g,
    float* __restrict__ out,
    int H, int W)
{
    constexpr int TS  = 64;            // output tile (TS x TS)
    constexpr int XSW = 72, XSH = 68;  // staged x tile: rows [-2,66), cols [-4,68)
    constexpr int E1W = 68, E1H = 66;  // e1 tile: rows [-1,65), cols [-2,66)

    __shared__ float xs [XSH * XSW];
    __shared__ float e1s[E1H * E1W];

    const int tid = threadIdx.x;
    const int tx0 = blockIdx.x * TS;
    const int ty0 = blockIdx.y * TS;
    const long long plane = (long long)H * (long long)W;
    const float* xb = x   + (long long)blockIdx.z * plane;
    float*       ob = out + (long long)blockIdx.z * plane;

    // Broadcast the two 3x3 structuring elements into registers.
    float s1[9], s2[9];
#pragma unroll
    for (int k = 0; k < 9; ++k) { s1[k] = s1g[k]; s2[k] = s2g[k]; }

    // ---- Stage x tile (with halo) into LDS, OOB filled with -10000 ----
    // 68 rows x 18 aligned float4 columns = 1224 quads, 256 threads.
    for (int idx = tid; idx < XSH * (XSW / 4); idx += 256) {
        int a  = idx / (XSW / 4);
        int qb = idx % (XSW / 4);
        int gy = ty0 - 2 + a;
        int gx = tx0 - 4 + qb * 4;
        int l  = a * XSW + qb * 4;
        bool quad_in = (gy >= 0) && (gy < H) && (gx >= 0) && (gx + 3 < W);
        bool aligned = ((((long long)gy * W + gx) & 3LL) == 0);
        if (quad_in && aligned) {
            const float* gsrc = xb + (long long)gy * W + gx;
#if defined(HAVE_ASYNC_LDS)
            v4i_t* gsrc_v = (v4i_t*)(void*)const_cast<float*>(gsrc);
            v4i_t* lds_v  = (v4i_t*)(void*)&xs[l];
            __builtin_amdgcn_global_load_async_to_lds_b128(
                (gv4_p)gsrc_v, (lv4_p)lds_v, /*offset=*/0, /*cpol=*/0);
#else
            *reinterpret_cast<float4*>(&xs[l]) =
                *reinterpret_cast<const float4*>(gsrc);
#endif
        } else {
#pragma unroll
            for (int o = 0; o < 4; ++o) {
                int gxx = gx + o;
                float v = NEGV;
                if (gy >= 0 && gy < H && gxx >= 0 && gxx < W)
                    v = xb[(long long)gy * W + gxx];
                xs[l + o] = v;
            }
        }
    }
    wait_async_zero();      // drain ASYNCcnt before making LDS visible
    __syncthreads();

    // ---- Stage 1: e1 = erosion(x, s1) over tile rows [-1,65) x cols [-2,66).
    // Positions outside the image are forced to exactly -10000 (second pad).
    for (int idx = tid; idx < E1H * (E1W / 4); idx += 256) {   // 66*17 = 1122
        int a  = idx / (E1W / 4);      // 0..65  (e1 row = a-1)
        int qb = idx % (E1W / 4);      // 0..16
        int i  = a - 1;
        int j0 = qb * 4 - 2;
        int gy = ty0 + i;
        float r0[6], r1[6], r2[6];
#pragma unroll
        for (int c = 0; c < 6; ++c) {
            int bcol = j0 + 3 + c;     // xs col for x(:, tx0+j0+c-1)
            r0[c] = xs[(a + 0) * XSW + bcol];
            r1[c] = xs[(a + 1) * XSW + bcol];
            r2[c] = xs[(a + 2) * XSW + bcol];
        }
        bool rin = (gy >= 0) && (gy < H);
#pragma unroll
        for (int o = 0; o < 4; ++o) {
            int gx = tx0 + j0 + o;
            float m =       r0[o + 0] - s1[0];
            m = fminf(m,    r0[o + 1] - s1[1]);
            m = fminf(m,    r0[o + 2] - s1[2]);
            m = fminf(m,    r1[o + 0] - s1[3]);
            m = fminf(m,    r1[o + 1] - s1[4]);
            m = fminf(m,    r1[o + 2] - s1[5]);
            m = fminf(m,    r2[o + 0] - s1[6]);
            m = fminf(m,    r2[o + 1] - s1[7]);
            m = fminf(m,    r2[o + 2] - s1[8]);
            if (!(rin && gx >= 0 && gx < W)) m = NEGV;  // exact border pad
            e1s[a * E1W + (j0 + 2 + o)] = m;
        }
    }
    __syncthreads();

    // ---- Stage 2: out = erosion(e1, s2), 64x64 outputs, float4 stores ----
    for (int idx = tid; idx < TS * (TS / 4); idx += 256) {     // 64*16 = 1024
        int i  = idx / (TS / 4);
        int qb = idx % (TS / 4);
        int j0 = qb * 4;
        float r0[6], r1[6], r2[6];
#pragma unroll
        for (int c = 0; c < 6; ++c) {
            int bcol = j0 + 1 + c;
            r0[c] = e1s[(i + 0) * E1W + bcol];
            r1[c] = e1s[(i + 1) * E1W + bcol];
            r2[c] = e1s[(i + 2) * E1W + bcol];
        }
        float o4[4];
#pragma unroll
        for (int o = 0; o < 4; ++o) {
            float m =       r0[o + 0] - s2[0];
            m = fminf(m,    r0[o + 1] - s2[1]);
            m = fminf(m,    r0[o + 2] - s2[2]);
            m = fminf(m,    r1[o + 0] - s2[3]);
            m = fminf(m,    r1[o + 1] - s2[4]);
            m = fminf(m,    r1[o + 2] - s2[5]);
            m = fminf(m,    r2[o + 0] - s2[6]);
            m = fminf(m,    r2[o + 1] - s2[7]);
            m = fminf(m,    r2[o + 2] - s2[8]);
            o4[o] = m;
        }
        int gy = ty0 + i;
        int gx = tx0 + j0;
        if (gy < H) {
            long long off = (long long)gy * W + gx;
            if (gx + 3 < W && ((off & 3LL) == 0)) {
                float4 v; v.x = o4[0]; v.y = o4[1]; v.z = o4[2]; v.w = o4[3];
                *reinterpret_cast<float4*>(&ob[off]) = v;
            } else {
#pragma unroll
                for (int o = 0; o < 4; ++o)
                    if (gx + o < W) ob[off + o] = o4[o];
            }
        }
    }
}

extern "C" void kernel_launch(void* const* d_in, const int* in_sizes, int n_in,
                              void* d_out, int out_size, void* d_ws, size_t ws_size,
                              hipStream_t stream)
{
    (void)n_in; (void)out_size; (void)d_ws; (void)ws_size;
    const float* x  = (const float*)d_in[0];
    const float* s1 = (const float*)d_in[1];
    const float* s2 = (const float*)d_in[2];
    float* out = (float*)d_out;

    const int H = 2048, W = 2048;
    int B = in_sizes[0] / (H * W);
    if (B < 1) B = 1;

    dim3 grid((W + 63) / 64, (H + 63) / 64, B);
    erode2_fused<<<grid, 256, 0, stream>>>(x, s1, s2, out, H, W);
}